// RelativeMultiHeadAttention_42468636623540
// MI455X (gfx1250) — compile-verified
//
#include <hip/hip_runtime.h>

// ---------------------------------------------------------------------------
// Relative multi-head attention (Transformer-XL style) for gfx1250 / MI455X.
// fp32 end-to-end; all matmuls on V_WMMA_F32_16X16X4_F32 (wave32).
//
// rel_shift algebra: shifted[b,qi,kj,n] = (q[c-1,kj,n]+v_bias)·pos[c-1,jj,n]
//   with h = 8*qi + b + 1024, jj = h/9, c = h%9 (row is zero when c==0).
// c depends only on (qi mod 9, b) -> tile queries by residue class mod 9 so
// the pos term is a plain WMMA GEMM fused into a flash-attention loop.
//
// All WMMA B-operand tiles are stored transposed in LDS so each fragment is
// one ds_load_b64 into an even-aligned VGPR pair (no v_dual_mov repacking).
// ---------------------------------------------------------------------------

#define D_MODEL 512
#define N_HEADS 8
#define D_HEAD  64
#define S_LEN   1024
#define B_SZ    8
#define M_TOT   (B_SZ * S_LEN)        // 8192 rows
#define ATT_SCALE 0.125f              // 1/sqrt(64)
#define NEG_MAX  -3.4028234663852886e38f

typedef float v2f __attribute__((ext_vector_type(2)));
typedef float v8f __attribute__((ext_vector_type(8)));

static __device__ __forceinline__ v8f wmma4(v2f a, v2f b, v8f c) {
  // D = A(16x4) * B(4x16) + C(16x16), fp32
  return __builtin_amdgcn_wmma_f32_16x16x4_f32(false, a, false, b,
                                               (short)0, c, false, false);
}

// ---------------------------------------------------------------------------
// Kernel 1: LayerNorm rows of (B*S, 512).  One block (256 thr) per row.
// ---------------------------------------------------------------------------
__global__ __launch_bounds__(256) void ln_kernel(
    const float* __restrict__ in, const float* __restrict__ gamma,
    const float* __restrict__ beta, float* __restrict__ xo) {
  const int row = blockIdx.x;
  const int t = threadIdx.x;
  const float* rp = in + (size_t)row * D_MODEL;
  float a0 = rp[t], a1 = rp[t + 256];

  __shared__ float red[8];
  float s = a0 + a1;
  for (int off = 16; off; off >>= 1) s += __shfl_xor(s, off, 32);
  if ((t & 31) == 0) red[t >> 5] = s;
  __syncthreads();
  float tot = 0.f;
  for (int i = 0; i < 8; ++i) tot += red[i];
  const float mean = tot * (1.0f / (float)D_MODEL);
  const float d0 = a0 - mean, d1 = a1 - mean;
  float vs = d0 * d0 + d1 * d1;
  for (int off = 16; off; off >>= 1) vs += __shfl_xor(vs, off, 32);
  __syncthreads();
  if ((t & 31) == 0) red[t >> 5] = vs;
  __syncthreads();
  float vtot = 0.f;
  for (int i = 0; i < 8; ++i) vtot += red[i];
  const float rstd = rsqrtf(vtot * (1.0f / (float)D_MODEL) + 1e-5f);

  xo[(size_t)row * D_MODEL + t]       = d0 * rstd * gamma[t] + beta[t];
  xo[(size_t)row * D_MODEL + t + 256] = d1 * rstd * gamma[t + 256] + beta[t + 256];
}

// ---------------------------------------------------------------------------
// Kernel 2: C(8192x512) = A(8192x512) @ W(512x512) [+ bias] [+ resid]
// 64x64 tile per workgroup, 4 waves, each wave: 16(M)x64(N) via 4 C-frags.
// A tile row-major (stride 36); W tile stored TRANSPOSED Bt[n][k] (stride 34)
// so B fragments are contiguous ds_load_b64, no repack moves.
// ---------------------------------------------------------------------------
__global__ __launch_bounds__(128) void gemm512_kernel(
    const float* __restrict__ A, const float* __restrict__ W,
    const float* __restrict__ bias, const float* __restrict__ resid,
    float* __restrict__ C) {
  __shared__ float As[64][36];
  __shared__ float Bt[64][34];   // [n][k]

  const int tid = threadIdx.x;
  const int w = tid >> 5, lane = tid & 31;
  const int half = lane >> 4, l16 = lane & 15, kb = half * 2;
  const int m0 = blockIdx.x * 64, n0 = blockIdx.y * 64;

  v8f c[4];
  for (int jt = 0; jt < 4; ++jt)
    for (int r = 0; r < 8; ++r) c[jt][r] = 0.f;

  for (int k0 = 0; k0 < D_MODEL; k0 += 32) {
    for (int i = tid; i < 512; i += 128) {            // A: 64x32
      int r = i >> 3, cc = (i & 7) << 2;
      *(float4*)&As[r][cc] =
          *(const float4*)(A + (size_t)(m0 + r) * D_MODEL + k0 + cc);
    }
    for (int i = tid; i < 512; i += 128) {            // W: 32x64 -> Bt 64x32
      int r = i >> 4, cc = (i & 15) << 2;
      float4 w4 = *(const float4*)(W + (size_t)(k0 + r) * D_MODEL + n0 + cc);
      Bt[cc + 0][r] = w4.x;
      Bt[cc + 1][r] = w4.y;
      Bt[cc + 2][r] = w4.z;
      Bt[cc + 3][r] = w4.w;
    }
    __syncthreads();
    const int arow = w * 16 + l16;
    for (int kk = 0; kk < 32; kk += 4) {
      v2f a = *(const v2f*)&As[arow][kk + kb];
#pragma unroll
      for (int jt = 0; jt < 4; ++jt) {
        v2f b = *(const v2f*)&Bt[jt * 16 + l16][kk + kb];
        c[jt] = wmma4(a, b, c[jt]);
      }
    }
    __syncthreads();
  }

  for (int jt = 0; jt < 4; ++jt) {
    const int col = n0 + jt * 16 + l16;
    const float bv = bias ? bias[col] : 0.f;
#pragma unroll
    for (int r = 0; r < 8; ++r) {
      const int row = m0 + w * 16 + r + half * 8;
      float vv = c[jt][r] + bv;
      if (resid) vv += resid[(size_t)row * D_MODEL + col];
      C[(size_t)row * D_MODEL + col] = vv;
    }
  }
}

// ---------------------------------------------------------------------------
// Kernel 3: fused attention (content + rel-shifted pos scores, flash softmax,
// attn @ V).  Grid: (18 = 9 classes x 2 halves, N_HEADS, B).  128 threads:
// wave w handles 16 queries qi = rcls + 9*(hc*64 + w*16 + t).
// ---------------------------------------------------------------------------
__global__ __launch_bounds__(128) void relattn_kernel(
    const float* __restrict__ q, const float* __restrict__ k,
    const float* __restrict__ v, const float* __restrict__ pos,
    const float* __restrict__ mask, const float* __restrict__ ub,
    const float* __restrict__ vb, float* __restrict__ ctx) {
  __shared__ float Qu[64][68];   // (q + u_bias) rows for this class
  __shared__ float Pv[64][68];   // gathered pos rows (rel_shift source)
  __shared__ float Ks[64][68];   // K tile   [j][d]   (B frag contiguous in d)
  __shared__ float Qs[64][68];   // (q[b_src] + v_bias) tile  [j][d]
  __shared__ float Vt[64][68];   // V tile TRANSPOSED [d][j] for P@V B frags
  __shared__ float Ps[4][16][68];// per-wave prob tiles (C-layout -> A-layout)

  const int tid = threadIdx.x;
  const int w = tid >> 5, lane = tid & 31;
  const int half = lane >> 4, l16 = lane & 15, kb = half * 2;
  const int b = blockIdx.z, n = blockIdx.y;
  const int rcls = blockIdx.x >> 1;   // residue class mod 9
  const int hc = blockIdx.x & 1;      // which half of the class (64 rows)
  const int c = (8 * rcls + b + S_LEN) % 9;   // uniform over the workgroup
  const bool posv = (c != 0);
  const int bsrc = posv ? (c - 1) : 0;

  // ---- stage Qu (q+u_bias) and Pv (gathered pos rows) --------------------
  for (int i = tid; i < 64 * 16; i += 128) {
    const int wr = i >> 4, cc = (i & 15) << 2;
    const int qi = rcls + 9 * (hc * 64 + wr);
    float4 qv = make_float4(0.f, 0.f, 0.f, 0.f);
    float4 pv = make_float4(0.f, 0.f, 0.f, 0.f);
    if (qi < S_LEN) {
      float4 qq = *(const float4*)(q + ((size_t)(b * S_LEN + qi)) * D_MODEL +
                                   n * D_HEAD + cc);
      float4 uu = *(const float4*)(ub + n * D_HEAD + cc);
      qv.x = qq.x + uu.x; qv.y = qq.y + uu.y;
      qv.z = qq.z + uu.z; qv.w = qq.w + uu.w;
      if (posv) {
        const int h = qi * 8 + b + S_LEN;
        const int jj = h / 9;
        pv = *(const float4*)(pos + ((size_t)(bsrc * S_LEN + jj)) * D_MODEL +
                              n * D_HEAD + cc);
      }
    }
    *(float4*)&Qu[wr][cc] = qv;
    *(float4*)&Pv[wr][cc] = pv;
  }

  float mrun[8], lrun[8];
  v8f acc[4];
  for (int r = 0; r < 8; ++r) { mrun[r] = NEG_MAX; lrun[r] = 0.f; }
  for (int nt = 0; nt < 4; ++nt)
    for (int r = 0; r < 8; ++r) acc[nt][r] = 0.f;

  for (int kj = 0; kj < S_LEN; kj += 64) {
    // ---- stage K, V(transposed), Q[b_src]+v_bias tiles -------------------
    for (int i = tid; i < 64 * 16; i += 128) {
      const int j = i >> 4, cc = (i & 15) << 2;
      const size_t kro =
          ((size_t)(b * S_LEN + kj + j)) * D_MODEL + n * D_HEAD + cc;
      *(float4*)&Ks[j][cc] = *(const float4*)(k + kro);
      float4 v4 = *(const float4*)(v + kro);
      Vt[cc + 0][j] = v4.x;
      Vt[cc + 1][j] = v4.y;
      Vt[cc + 2][j] = v4.z;
      Vt[cc + 3][j] = v4.w;
      float4 qs = make_float4(0.f, 0.f, 0.f, 0.f);
      if (posv) {
        float4 qq = *(const float4*)(
            q + ((size_t)(bsrc * S_LEN + kj + j)) * D_MODEL + n * D_HEAD + cc);
        float4 vv = *(const float4*)(vb + n * D_HEAD + cc);
        qs.x = qq.x + vv.x; qs.y = qq.y + vv.y;
        qs.z = qq.z + vv.z; qs.w = qq.w + vv.w;
      }
      *(float4*)&Qs[j][cc] = qs;
    }
    __syncthreads();

    // ---- scores: content + pos (two WMMA chains) -------------------------
    v8f s[4];
    for (int jt = 0; jt < 4; ++jt)
      for (int r = 0; r < 8; ++r) s[jt][r] = 0.f;
    const int arow = w * 16 + l16;
    for (int kk = 0; kk < 64; kk += 4) {
      v2f a  = *(const v2f*)&Qu[arow][kk + kb];
#pragma unroll
      for (int jt = 0; jt < 4; ++jt) {
        v2f bb = *(const v2f*)&Ks[jt * 16 + l16][kk + kb];
        s[jt] = wmma4(a, bb, s[jt]);
      }
    }
    if (posv) {
      for (int kk = 0; kk < 64; kk += 4) {
        v2f ap = *(const v2f*)&Pv[arow][kk + kb];
#pragma unroll
        for (int jt = 0; jt < 4; ++jt) {
          v2f bb = *(const v2f*)&Qs[jt * 16 + l16][kk + kb];
          s[jt] = wmma4(ap, bb, s[jt]);
        }
      }
    }

    // ---- scale + mask ----------------------------------------------------
#pragma unroll
    for (int jt = 0; jt < 4; ++jt)
#pragma unroll
      for (int r = 0; r < 8; ++r) {
        float sv = s[jt][r] * ATT_SCALE;
        const int m = w * 16 + r + half * 8;
        const int qi = rcls + 9 * (hc * 64 + m);
        if (qi < S_LEN) {
          const float mv =
              mask[((size_t)b * S_LEN + qi) * S_LEN + kj + jt * 16 + l16];
          if (mv == 0.0f) sv = NEG_MAX;
        }
        s[jt][r] = sv;
      }

    // ---- online softmax (row stats via 16-lane xor shuffles) -------------
#pragma unroll
    for (int r = 0; r < 8; ++r) {
      float mx = fmaxf(fmaxf(s[0][r], s[1][r]), fmaxf(s[2][r], s[3][r]));
      for (int off = 1; off < 16; off <<= 1)
        mx = fmaxf(mx, __shfl_xor(mx, off, 32));
      const float mnew = fmaxf(mrun[r], mx);
      const float corr = __expf(mrun[r] - mnew);
      float pj[4], psum = 0.f;
#pragma unroll
      for (int jt = 0; jt < 4; ++jt) {
        pj[jt] = __expf(s[jt][r] - mnew);
        psum += pj[jt];
      }
      for (int off = 1; off < 16; off <<= 1)
        psum += __shfl_xor(psum, off, 32);
      lrun[r] = lrun[r] * corr + psum;
      mrun[r] = mnew;
#pragma unroll
      for (int nt = 0; nt < 4; ++nt) acc[nt][r] *= corr;
      const int m = r + half * 8;
#pragma unroll
      for (int jt = 0; jt < 4; ++jt) Ps[w][m][jt * 16 + l16] = pj[jt];
    }

    // ---- acc += P(16x64) @ V(64x64)  (wave-local LDS round trip) ---------
    for (int kk = 0; kk < 64; kk += 4) {
      v2f a = *(const v2f*)&Ps[w][l16][kk + kb];
#pragma unroll
      for (int nt = 0; nt < 4; ++nt) {
        v2f bb = *(const v2f*)&Vt[nt * 16 + l16][kk + kb];
        acc[nt] = wmma4(a, bb, acc[nt]);
      }
    }
    __syncthreads();
  }

  // ---- normalize + store ctx ---------------------------------------------
#pragma unroll
  for (int r = 0; r < 8; ++r) {
    const int m = w * 16 + r + half * 8;
    const int qi = rcls + 9 * (hc * 64 + m);
    if (qi < S_LEN) {
      const float inv = 1.0f / lrun[r];
#pragma unroll
      for (int nt = 0; nt < 4; ++nt)
        ctx[((size_t)(b * S_LEN + qi)) * D_MODEL + n * D_HEAD + nt * 16 + l16] =
            acc[nt][r] * inv;
    }
  }
}

// ---------------------------------------------------------------------------
extern "C" void kernel_launch(void* const* d_in, const int* in_sizes, int n_in,
                              void* d_out, int out_size, void* d_ws,
                              size_t ws_size, hipStream_t stream) {
  (void)in_sizes; (void)n_in; (void)out_size; (void)ws_size;
  const float* inputs = (const float*)d_in[0];
  const float* pos    = (const float*)d_in[1];
  const float* mask   = (const float*)d_in[2];
  const float* Wq     = (const float*)d_in[3];
  const float* bq     = (const float*)d_in[4];
  const float* Wk     = (const float*)d_in[5];
  const float* bk     = (const float*)d_in[6];
  const float* Wv     = (const float*)d_in[7];
  const float* bv     = (const float*)d_in[8];
  const float* Wo     = (const float*)d_in[9];
  const float* ub     = (const float*)d_in[10];
  const float* vbias  = (const float*)d_in[11];
  const float* gamma  = (const float*)d_in[12];
  const float* beta   = (const float*)d_in[13];
  float* out = (float*)d_out;

  const size_t mat = (size_t)M_TOT * D_MODEL;   // 4M floats = 16 MB
  float* x   = (float*)d_ws;
  float* q   = x + mat;
  float* kk  = q + mat;
  float* vv  = kk + mat;
  float* ctx = vv + mat;

  ln_kernel<<<M_TOT, 256, 0, stream>>>(inputs, gamma, beta, x);

  dim3 ggrid(M_TOT / 64, D_MODEL / 64);   // 128 x 8
  gemm512_kernel<<<ggrid, 128, 0, stream>>>(x, Wq, bq, nullptr, q);
  gemm512_kernel<<<ggrid, 128, 0, stream>>>(x, Wk, bk, nullptr, kk);
  gemm512_kernel<<<ggrid, 128, 0, stream>>>(x, Wv, bv, nullptr, vv);

  relattn_kernel<<<dim3(18, N_HEADS, B_SZ), 128, 0, stream>>>(
      q, kk, vv, pos, mask, ub, vbias, ctx);

  gemm512_kernel<<<ggrid, 128, 0, stream>>>(ctx, Wo, nullptr, x, out);
}